// SoftmaxN_73521250173290
// MI455X (gfx1250) — compile-verified
//
#include <hip/hip_runtime.h>
#include <stdint.h>

#define NEG_INF (-__builtin_inff())

// ---------------- wave32 shuffle reductions ----------------
__device__ __forceinline__ float wave_max(float v) {
#pragma unroll
  for (int s = 16; s > 0; s >>= 1) v = fmaxf(v, __shfl_xor(v, s, 32));
  return v;
}
__device__ __forceinline__ float wave_sum(float v) {
#pragma unroll
  for (int s = 16; s > 0; s >>= 1) v += __shfl_xor(v, s, 32);
  return v;
}

// ---------------- kernel 1: per-chunk scan of sizes ----------------
// One block scans 256 sizes -> exclusive offsets within chunk + chunk total.
__global__ __launch_bounds__(256) void scan_chunks(const int* __restrict__ sizes,
                                                   int* __restrict__ seg_excl,
                                                   int* __restrict__ chunk_tot,
                                                   int B) {
  __shared__ int tmp[256];
  const int tid = threadIdx.x;
  const int i = blockIdx.x * 256 + tid;
  const int v = (i < B) ? sizes[i] : 0;
  tmp[tid] = v;
  __syncthreads();
#pragma unroll
  for (int d = 1; d < 256; d <<= 1) {
    int t = (tid >= d) ? tmp[tid - d] : 0;
    __syncthreads();
    tmp[tid] += t;
    __syncthreads();
  }
  if (i < B) seg_excl[i] = tmp[tid] - v;   // exclusive within chunk
  if (tid == 255) chunk_tot[blockIdx.x] = tmp[255];
}

// ---------------- kernel 2: exclusive scan of chunk totals (in place) ----------------
__global__ __launch_bounds__(256) void scan_bases(int* __restrict__ chunk, int n) {
  __shared__ int tmp[256];
  __shared__ int carry;
  const int tid = threadIdx.x;
  if (tid == 0) carry = 0;
  __syncthreads();
  for (int base = 0; base < n; base += 256) {
    const int i = base + tid;
    const int v = (i < n) ? chunk[i] : 0;
    tmp[tid] = v;
    __syncthreads();
#pragma unroll
    for (int d = 1; d < 256; d <<= 1) {
      int t = (tid >= d) ? tmp[tid - d] : 0;
      __syncthreads();
      tmp[tid] += t;
      __syncthreads();
    }
    const int excl = tmp[tid] - v + carry;
    if (i < n) chunk[i] = excl;
    __syncthreads();
    if (tid == 0) carry += tmp[255];
    __syncthreads();
  }
}

// ---------------- kernel 3: one block per segment ----------------
// Async-stage the segment tile into LDS (CDNA5 ASYNCcnt path), then
// max-reduce, exp, sum-reduce, scale by n/sum, store once.
__global__ __launch_bounds__(256) void seg_softmax(const float* __restrict__ x,
                                                   const int* __restrict__ sizes,
                                                   const int* __restrict__ seg_excl,
                                                   const int* __restrict__ chunk_base,
                                                   float* __restrict__ out) {
  __shared__ float tile[512];
  __shared__ float smax[8];
  __shared__ float ssum[8];

  const int g   = blockIdx.x;
  const int tid = threadIdx.x;
  const int n   = sizes[g];                       // 128..512
  const int off = seg_excl[g] + chunk_base[g >> 8];
  const float* gseg = x + off;

  // ---- async global -> LDS staging, 16B packets per lane (ASYNCcnt path) ----
  const unsigned lds0 = (unsigned)(uintptr_t)&tile[0];  // low 32 bits = LDS byte offset
  const int nvec = n >> 2;                              // # of b128 packets (sizes % 4 == 0)
  if (tid < nvec) {
    unsigned dst  = lds0 + ((unsigned)tid << 4);
    unsigned voff = (unsigned)tid << 4;
    asm volatile("global_load_async_to_lds_b128 %0, %1, %2"
                 :: "v"(dst), "v"(voff), "s"(gseg) : "memory");
  }
  // scalar tail (never taken for multiple-of-4 sizes, kept for generality)
  for (int i = (nvec << 2) + tid; i < n; i += 256) {
    unsigned dst  = lds0 + ((unsigned)i << 2);
    unsigned voff = (unsigned)i << 2;
    asm volatile("global_load_async_to_lds_b32 %0, %1, %2"
                 :: "v"(dst), "v"(voff), "s"(gseg) : "memory");
  }
  asm volatile("s_wait_asynccnt 0" ::: "memory");  // own wave's async writes done
  __syncthreads();                                  // all waves' writes visible

  // ---- pass 1: block max ----
  const float v0 = (tid < n)       ? tile[tid]       : NEG_INF;
  const float v1 = (tid + 256 < n) ? tile[tid + 256] : NEG_INF;
  float m = wave_max(fmaxf(v0, v1));
  const int lane = tid & 31, wv = tid >> 5;
  if (lane == 0) smax[wv] = m;
  __syncthreads();
  const float bm = fmaxf(fmaxf(fmaxf(smax[0], smax[1]), fmaxf(smax[2], smax[3])),
                         fmaxf(fmaxf(smax[4], smax[5]), fmaxf(smax[6], smax[7])));

  // ---- pass 2: exp + block sum (accurate expf; cost hidden under HBM stream) ----
  const float e0 = (tid < n)       ? expf(v0 - bm) : 0.0f;
  const float e1 = (tid + 256 < n) ? expf(v1 - bm) : 0.0f;
  float s = wave_sum(e0 + e1);
  if (lane == 0) ssum[wv] = s;
  __syncthreads();
  const float bs = ((ssum[0] + ssum[1]) + (ssum[2] + ssum[3])) +
                   ((ssum[4] + ssum[5]) + (ssum[6] + ssum[7]));

  const float scale = (float)n / bs;
  if (tid < n)       out[off + tid]       = e0 * scale;
  if (tid + 256 < n) out[off + tid + 256] = e1 * scale;
}

// ---------------- host launcher ----------------
extern "C" void kernel_launch(void* const* d_in, const int* in_sizes, int n_in,
                              void* d_out, int out_size, void* d_ws, size_t ws_size,
                              hipStream_t stream) {
  const float* x     = (const float*)d_in[0];
  const int*   sizes = (const int*)d_in[1];
  // d_in[2] (segment_ids) is not needed: offsets are derived from `sizes`.
  float* out = (float*)d_out;

  const int B = in_sizes[1];
  const int nchunk = (B + 255) / 256;

  int* seg_excl   = (int*)d_ws;          // [B]      exclusive offset within chunk
  int* chunk_base = seg_excl + B;        // [nchunk] chunk totals -> exclusive bases

  scan_chunks<<<nchunk, 256, 0, stream>>>(sizes, seg_excl, chunk_base, B);
  scan_bases<<<1, 256, 0, stream>>>(chunk_base, nchunk);
  seg_softmax<<<B, 256, 0, stream>>>(x, sizes, seg_excl, chunk_base, out);
}